// GRIT_74466142978757
// MI455X (gfx1250) — compile-verified
//
#include <hip/hip_runtime.h>
#include <hip/hip_bf16.h>
#include <math.h>

// ---------------- problem constants (match reference) ----------------
#define NNODES 2048
#define NGRAPH 16
#define NPG    128
#define HID    256
#define NH     8
#define HD     32
#define NLAYER 4
#define NEDGE  32768
#define MAXDEG 100
#define MLPH   128   // HIDDEN/2

typedef float v2f __attribute__((ext_vector_type(2)));
typedef float v8f __attribute__((ext_vector_type(8)));

// f32 WMMA: D(16x16) = A(16x4) * B(4x16) + C
__device__ __forceinline__ v8f wmma4(v2f a, v2f b, v8f c) {
  return __builtin_amdgcn_wmma_f32_16x16x4_f32(
      /*neg_a=*/false, a, /*neg_b=*/false, b,
      /*c_mod=*/(short)0, c, /*reuse_a=*/false, /*reuse_b=*/false);
}

__device__ __forceinline__ float wave_max(float v) {
  #pragma unroll
  for (int o = 16; o > 0; o >>= 1) v = fmaxf(v, __shfl_xor(v, o, 32));
  return v;
}
__device__ __forceinline__ float wave_sum(float v) {
  #pragma unroll
  for (int o = 16; o > 0; o >>= 1) v += __shfl_xor(v, o, 32);
  return v;
}

// ---- 32x32 super-tile micro-kernel: 2x2 accumulators, 4 WMMAs per k-step ----
// A row-major [.. x lda], B row-major [K x ldb].
// acc[0]=(m0,n0) acc[1]=(m0,n1) acc[2]=(m1,n0) acc[3]=(m1,n1)
__device__ __forceinline__ void gemm32x32(const float* __restrict__ A, int lda,
                                          const float* __restrict__ B, int ldb,
                                          int K, int tm, int tn, int lane,
                                          v8f acc[4]) {
  int half = lane >> 4, l = lane & 15;
  const float* arow0 = A + (size_t)(tm + l) * lda;
  const float* arow1 = A + (size_t)(tm + 16 + l) * lda;
  for (int k0 = 0; k0 < K; k0 += 4) {
    int ka = k0 + (half << 1);
    v2f a0, a1, b0, b1;
    a0.x = arow0[ka];      a0.y = arow0[ka + 1];      // contiguous -> b64
    a1.x = arow1[ka];      a1.y = arow1[ka + 1];
    const float* br0 = B + (size_t)ka * ldb;
    const float* br1 = B + (size_t)(ka + 1) * ldb;
    b0.x = br0[tn + l];        b0.y = br1[tn + l];
    b1.x = br0[tn + 16 + l];   b1.y = br1[tn + 16 + l];
    acc[0] = wmma4(a0, b0, acc[0]);
    acc[1] = wmma4(a0, b1, acc[1]);
    acc[2] = wmma4(a1, b0, acc[2]);
    acc[3] = wmma4(a1, b1, acc[3]);
  }
}

// ---------------- embedding gather: x = node_emb[x_idx] ----------------
__global__ __launch_bounds__(HID)
void k_embed(const int* __restrict__ x_idx, const float* __restrict__ node_emb,
             float* __restrict__ x) {
  int i = blockIdx.x, d = threadIdx.x;
  x[i * HID + d] = node_emb[x_idx[i] * HID + d];
}

// ---------------- degrees: deg[src] += 1 ----------------
__global__ __launch_bounds__(256)
void k_degree(const int* __restrict__ src, int* __restrict__ deg) {
  int e = blockIdx.x * 256 + threadIdx.x;
  if (e < NEDGE) atomicAdd(&deg[src[e]], 1);
}

// ---------------- scatter T (random-walk) and EB (edge bias) ----------------
__global__ __launch_bounds__(256)
void k_scatter(const int* __restrict__ src, const int* __restrict__ dst,
               const int* __restrict__ eattr, const int* __restrict__ batch,
               const int* __restrict__ deg, const float* __restrict__ edge_emb,
               float* __restrict__ T, float* __restrict__ EB) {
  int e = blockIdx.x * 256 + threadIdx.x;
  if (e >= NEDGE) return;
  int s = src[e], d = dst[e];
  if (batch[s] != batch[d]) return;   // cross-graph edges contribute nothing
  int g = batch[s];
  int u = s & (NPG - 1), v = d & (NPG - 1);
  float idu = 1.0f / fmaxf((float)deg[s], 1.0f);
  float idv = 1.0f / fmaxf((float)deg[d], 1.0f);
  atomicAdd(&T[(g * NPG + u) * NPG + v], idu);
  atomicAdd(&T[(g * NPG + v) * NPG + u], idv);
  const float* ev = edge_emb + eattr[e] * NH;
  float* p1 = EB + (size_t)((g * NPG + u) * NPG + v) * NH;
  float* p2 = EB + (size_t)((g * NPG + v) * NPG + u) * NH;
  #pragma unroll
  for (int h = 0; h < NH; ++h) {
    float t = ev[h];
    atomicAdd(&p1[h], t);
    atomicAdd(&p2[h], t);
  }
}

// ---------------- batched 128x128x128 GEMM per graph (walk powers) ----------------
// grid = (2, NGRAPH), block = 256 (8 waves, one 32x32 super-tile each; 16/graph)
__global__ __launch_bounds__(256)
void k_bgemm128(const float* __restrict__ A, const float* __restrict__ B,
                float* __restrict__ C) {
  int g = blockIdx.y;
  const float* Ag = A + (size_t)g * NPG * NPG;
  const float* Bg = B + (size_t)g * NPG * NPG;
  float*       Cg = C + (size_t)g * NPG * NPG;
  int lane = threadIdx.x & 31, wave = threadIdx.x >> 5;
  int tile = blockIdx.x * 8 + wave;          // 0..15 (4x4 super-tiles)
  int tm = (tile >> 2) << 5, tn = (tile & 3) << 5;
  int half = lane >> 4, l = lane & 15;
  v8f acc[4] = {{0.f,0.f,0.f,0.f,0.f,0.f,0.f,0.f},{0.f,0.f,0.f,0.f,0.f,0.f,0.f,0.f},
                {0.f,0.f,0.f,0.f,0.f,0.f,0.f,0.f},{0.f,0.f,0.f,0.f,0.f,0.f,0.f,0.f}};
  gemm32x32(Ag, NPG, Bg, NPG, NPG, tm, tn, lane, acc);
  #pragma unroll
  for (int t = 0; t < 4; ++t) {
    int dm = (t >> 1) << 4, dn = (t & 1) << 4;
    #pragma unroll
    for (int r = 0; r < 8; ++r)
      Cg[(tm + dm + (half << 3) + r) * NPG + tn + dn + l] = acc[t][r];
  }
}

// ---------------- generic GEMM: C[M,N] = A[M,K] @ W[K,N] + bias (opt relu) ----------------
// block = 256 (8 waves, one 32x32 super-tile each); grid.x = (M/32 * N/32) / 8
__global__ __launch_bounds__(256)
void k_gemm(const float* __restrict__ A, const float* __restrict__ W,
            const float* __restrict__ bias, float* __restrict__ C,
            int M, int K, int N, int relu) {
  int lane = threadIdx.x & 31, wave = threadIdx.x >> 5;
  int sn = N >> 5;
  int tile = blockIdx.x * 8 + wave;
  int tm = (tile / sn) << 5, tn = (tile % sn) << 5;
  int half = lane >> 4, l = lane & 15;
  v8f acc[4] = {{0.f,0.f,0.f,0.f,0.f,0.f,0.f,0.f},{0.f,0.f,0.f,0.f,0.f,0.f,0.f,0.f},
                {0.f,0.f,0.f,0.f,0.f,0.f,0.f,0.f},{0.f,0.f,0.f,0.f,0.f,0.f,0.f,0.f}};
  gemm32x32(A, K, W, N, K, tm, tn, lane, acc);
  #pragma unroll
  for (int t = 0; t < 4; ++t) {
    int dm = (t >> 1) << 4, dn = (t & 1) << 4;
    float bb = bias[tn + dn + l];
    #pragma unroll
    for (int r = 0; r < 8; ++r) {
      float vv = acc[t][r] + bb;
      if (relu) vv = fmaxf(vv, 0.0f);
      C[(size_t)(tm + dm + (half << 3) + r) * N + tn + dn + l] = vv;
    }
  }
}

// ---------------- RPE MLP + edge bias -> attn bias (B,H,n,n) ----------------
__global__ __launch_bounds__(256)
void k_rpe(const float* __restrict__ T1, const float* __restrict__ T2,
           const float* __restrict__ T3, const float* __restrict__ T4,
           const float* __restrict__ W1, const float* __restrict__ b1,
           const float* __restrict__ W2, const float* __restrict__ b2,
           const float* __restrict__ EB, float* __restrict__ bias) {
  __shared__ float sW1[5 * MLPH];
  __shared__ float sb1[MLPH];
  __shared__ float sW2[MLPH * NH];
  __shared__ float sb2[NH];
  for (int t = threadIdx.x; t < 5 * MLPH; t += 256) sW1[t] = W1[t];
  for (int t = threadIdx.x; t < MLPH;     t += 256) sb1[t] = b1[t];
  for (int t = threadIdx.x; t < MLPH*NH;  t += 256) sW2[t] = W2[t];
  if (threadIdx.x < NH) sb2[threadIdx.x] = b2[threadIdx.x];
  __syncthreads();

  int idx = blockIdx.x * 256 + threadIdx.x;          // (g*128+i)*128 + j
  int j = idx & (NPG - 1);
  int gi = idx >> 7;
  int i = gi & (NPG - 1);
  int g = gi >> 7;

  float w0 = (i == j) ? 1.0f : 0.0f;
  float w1v = T1[idx], w2v = T2[idx], w3v = T3[idx], w4v = T4[idx];

  float out[NH];
  #pragma unroll
  for (int h = 0; h < NH; ++h) out[h] = sb2[h];

  for (int m = 0; m < MLPH; ++m) {
    float hsum = sb1[m]
               + w0  * sW1[0 * MLPH + m]
               + w1v * sW1[1 * MLPH + m]
               + w2v * sW1[2 * MLPH + m]
               + w3v * sW1[3 * MLPH + m]
               + w4v * sW1[4 * MLPH + m];
    hsum = fmaxf(hsum, 0.0f);
    #pragma unroll
    for (int h = 0; h < NH; ++h) out[h] += hsum * sW2[m * NH + h];
  }
  #pragma unroll
  for (int h = 0; h < NH; ++h)
    bias[((size_t)(g * NH + h) * NPG + i) * NPG + j] = out[h] + EB[(size_t)idx * NH + h];
}

// ---------------- x += degree_emb[min(deg,100)] ----------------
__global__ __launch_bounds__(HID)
void k_adddeg(float* __restrict__ x, const int* __restrict__ deg,
              const float* __restrict__ degree_emb) {
  int i = blockIdx.x, d = threadIdx.x;
  int dc = min(deg[i], MAXDEG);
  x[i * HID + d] += degree_emb[dc * HID + d];
}

// ---------------- scores: S[bh,i,j] = scale*Q.K^T + bias, per (g,h) block ----------------
// grid = (2, NGRAPH*NH), block = 256 (8 waves -> 16 super-tiles of 32x32)
// Both Q and K^T fragments are contiguous per lane (b64 loads).
__global__ __launch_bounds__(256)
void k_scores(const float* __restrict__ Q, const float* __restrict__ Kt,
              const float* __restrict__ bias, float* __restrict__ S, float scale) {
  int bh = blockIdx.y;
  int g = bh >> 3, h = bh & 7;
  int lane = threadIdx.x & 31, wave = threadIdx.x >> 5;
  int tile = blockIdx.x * 8 + wave;          // 0..15 (4x4 super-tiles)
  int tm = (tile >> 2) << 5, tn = (tile & 3) << 5;
  int half = lane >> 4, l = lane & 15;
  const float* Qb = Q  + (size_t)g * NPG * HID + h * HD;   // row i -> Qb[i*HID + d]
  const float* Kb = Kt + (size_t)g * NPG * HID + h * HD;   // row j -> Kb[j*HID + d]
  const float* aq0 = Qb + (size_t)(tm + l) * HID;
  const float* aq1 = Qb + (size_t)(tm + 16 + l) * HID;
  const float* bk0 = Kb + (size_t)(tn + l) * HID;
  const float* bk1 = Kb + (size_t)(tn + 16 + l) * HID;
  v8f acc[4] = {{0.f,0.f,0.f,0.f,0.f,0.f,0.f,0.f},{0.f,0.f,0.f,0.f,0.f,0.f,0.f,0.f},
                {0.f,0.f,0.f,0.f,0.f,0.f,0.f,0.f},{0.f,0.f,0.f,0.f,0.f,0.f,0.f,0.f}};
  #pragma unroll
  for (int k0 = 0; k0 < HD; k0 += 4) {
    int ka = k0 + (half << 1);
    v2f a0, a1, b0, b1;
    a0.x = aq0[ka]; a0.y = aq0[ka + 1];
    a1.x = aq1[ka]; a1.y = aq1[ka + 1];
    b0.x = bk0[ka]; b0.y = bk0[ka + 1];
    b1.x = bk1[ka]; b1.y = bk1[ka + 1];
    acc[0] = wmma4(a0, b0, acc[0]);
    acc[1] = wmma4(a0, b1, acc[1]);
    acc[2] = wmma4(a1, b0, acc[2]);
    acc[3] = wmma4(a1, b1, acc[3]);
  }
  #pragma unroll
  for (int t = 0; t < 4; ++t) {
    int dm = (t >> 1) << 4, dn = (t & 1) << 4;
    #pragma unroll
    for (int r = 0; r < 8; ++r) {
      size_t o = ((size_t)bh * NPG + tm + dm + (half << 3) + r) * NPG + tn + dn + l;
      S[o] = acc[t][r] * scale + bias[o];
    }
  }
}

// ---------------- row softmax over 128 cols; one wave per row ----------------
__global__ __launch_bounds__(256)
void k_softmax(float* __restrict__ S) {
  int row = blockIdx.x * 8 + (threadIdx.x >> 5);
  int lane = threadIdx.x & 31;
  float* p = S + (size_t)row * NPG;
  float v[4];
  float m = -1e30f;
  #pragma unroll
  for (int t = 0; t < 4; ++t) { v[t] = p[lane + 32 * t]; m = fmaxf(m, v[t]); }
  m = wave_max(m);
  float s = 0.0f;
  #pragma unroll
  for (int t = 0; t < 4; ++t) { v[t] = __expf(v[t] - m); s += v[t]; }
  s = wave_sum(s);
  float inv = 1.0f / s;
  #pragma unroll
  for (int t = 0; t < 4; ++t) p[lane + 32 * t] = v[t] * inv;
}

// ---------------- out[i,hd] = sum_j AW[i,j] * V[j,hd], per (g,h) ----------------
// grid = (1, NGRAPH*NH), block = 128 (4 waves -> 4 super-tiles of 32x32)
__global__ __launch_bounds__(128)
void k_av(const float* __restrict__ S, const float* __restrict__ V,
          float* __restrict__ O) {
  int bh = blockIdx.y;
  int g = bh >> 3, h = bh & 7;
  int lane = threadIdx.x & 31, wave = threadIdx.x >> 5;
  int tm = wave << 5, tn = 0;                // 4 super-tiles stacked in M
  int half = lane >> 4, l = lane & 15;
  const float* Ab = S + (size_t)bh * NPG * NPG;        // 128x128, lda = NPG
  const float* Bb = V + (size_t)g * NPG * HID + h * HD; // B[k][n] = Bb[k*HID + n]
  v8f acc[4] = {{0.f,0.f,0.f,0.f,0.f,0.f,0.f,0.f},{0.f,0.f,0.f,0.f,0.f,0.f,0.f,0.f},
                {0.f,0.f,0.f,0.f,0.f,0.f,0.f,0.f},{0.f,0.f,0.f,0.f,0.f,0.f,0.f,0.f}};
  gemm32x32(Ab, NPG, Bb, HID, NPG, tm, tn, lane, acc);
  #pragma unroll
  for (int t = 0; t < 4; ++t) {
    int dm = (t >> 1) << 4, dn = (t & 1) << 4;
    #pragma unroll
    for (int r = 0; r < 8; ++r)
      O[(size_t)(g * NPG + tm + dm + (half << 3) + r) * HID + h * HD + tn + dn + l] =
          acc[t][r];
  }
}

// ---------------- layernorm(x + delta) * s + b -> x (in place) ----------------
__global__ __launch_bounds__(256)
void k_ln(float* __restrict__ x, const float* __restrict__ delta,
          const float* __restrict__ s, const float* __restrict__ b) {
  int row = blockIdx.x * 8 + (threadIdx.x >> 5);
  int lane = threadIdx.x & 31;
  float v[8];
  float sum = 0.0f;
  #pragma unroll
  for (int t = 0; t < 8; ++t) {
    int c = lane + 32 * t;
    v[t] = x[(size_t)row * HID + c] + delta[(size_t)row * HID + c];
    sum += v[t];
  }
  float mean = wave_sum(sum) * (1.0f / HID);
  float var = 0.0f;
  #pragma unroll
  for (int t = 0; t < 8; ++t) { float d = v[t] - mean; var += d * d; }
  var = wave_sum(var) * (1.0f / HID);
  float rstd = rsqrtf(var + 1e-5f);
  #pragma unroll
  for (int t = 0; t < 8; ++t) {
    int c = lane + 32 * t;
    x[(size_t)row * HID + c] = (v[t] - mean) * rstd * s[c] + b[c];
  }
}

// ---------------- global add pool: out[g,d] = sum_i x[g*128+i,d] ----------------
__global__ __launch_bounds__(HID)
void k_pool(const float* __restrict__ x, float* __restrict__ out) {
  int g = blockIdx.x, d = threadIdx.x;
  float s = 0.0f;
  for (int i = 0; i < NPG; ++i) s += x[(size_t)(g * NPG + i) * HID + d];
  out[g * HID + d] = s;
}

// ======================= host orchestration =======================
extern "C" void kernel_launch(void* const* d_in, const int* in_sizes, int n_in,
                              void* d_out, int out_size, void* d_ws, size_t ws_size,
                              hipStream_t stream) {
  (void)in_sizes; (void)n_in; (void)out_size; (void)ws_size;

  const int*   x_idx      = (const int*)d_in[0];
  const int*   edge_index = (const int*)d_in[1];
  const int*   edge_attr  = (const int*)d_in[2];
  const int*   batch      = (const int*)d_in[3];
  const float* node_emb   = (const float*)d_in[4];
  const float* edge_emb   = (const float*)d_in[5];
  const float* degree_emb = (const float*)d_in[6];
  const float* rpe_W1     = (const float*)d_in[7];
  const float* rpe_b1     = (const float*)d_in[8];
  const float* rpe_W2     = (const float*)d_in[9];
  const float* rpe_b2     = (const float*)d_in[10];
  const float* Wq = (const float*)d_in[11]; const float* bq = (const float*)d_in[12];
  const float* Wk = (const float*)d_in[13]; const float* bk = (const float*)d_in[14];
  const float* Wv = (const float*)d_in[15]; const float* bv = (const float*)d_in[16];
  const float* Wo = (const float*)d_in[17]; const float* bo = (const float*)d_in[18];
  const float* ln1_s = (const float*)d_in[19]; const float* ln1_b = (const float*)d_in[20];
  const float* ln2_s = (const float*)d_in[21]; const float* ln2_b = (const float*)d_in[22];
  const float* ffW1 = (const float*)d_in[23]; const float* ffb1 = (const float*)d_in[24];
  const float* ffW2 = (const float*)d_in[25]; const float* ffb2 = (const float*)d_in[26];

  const int* src = edge_index;
  const int* dst = edge_index + NEDGE;

  // ---- workspace layout ----
  char* p = (char*)d_ws;
  size_t off = 0;
  auto alloc = [&](size_t bytes) {
    void* r = p + off;
    off += (bytes + 255) & ~(size_t)255;
    return r;
  };
  int*   deg  = (int*)  alloc((size_t)NNODES * 4);
  float* x    = (float*)alloc((size_t)NNODES * HID * 4);
  float* T1   = (float*)alloc((size_t)NGRAPH * NPG * NPG * 4);
  float* T2   = (float*)alloc((size_t)NGRAPH * NPG * NPG * 4);
  float* T3   = (float*)alloc((size_t)NGRAPH * NPG * NPG * 4);
  float* T4   = (float*)alloc((size_t)NGRAPH * NPG * NPG * 4);
  float* EB   = (float*)alloc((size_t)NGRAPH * NPG * NPG * NH * 4);
  float* abias= (float*)alloc((size_t)NGRAPH * NH * NPG * NPG * 4);
  float* q    = (float*)alloc((size_t)NNODES * HID * 4);
  float* kk   = (float*)alloc((size_t)NNODES * HID * 4);
  float* vv   = (float*)alloc((size_t)NNODES * HID * 4);
  float* att  = (float*)alloc((size_t)NNODES * HID * 4);
  float* sc   = (float*)alloc((size_t)NGRAPH * NH * NPG * NPG * 4);
  float* ffh  = (float*)alloc((size_t)NNODES * 2 * HID * 4);
  float* proj = (float*)alloc((size_t)NNODES * HID * 4);

  // ---- zero scatter targets ----
  hipMemsetAsync(deg, 0, (size_t)NNODES * 4, stream);
  hipMemsetAsync(T1,  0, (size_t)NGRAPH * NPG * NPG * 4, stream);
  hipMemsetAsync(EB,  0, (size_t)NGRAPH * NPG * NPG * NH * 4, stream);

  // ---- graph preprocessing ----
  k_embed<<<NNODES, HID, 0, stream>>>(x_idx, node_emb, x);
  k_degree<<<NEDGE / 256, 256, 0, stream>>>(src, deg);
  k_scatter<<<NEDGE / 256, 256, 0, stream>>>(src, dst, edge_attr, batch, deg,
                                             edge_emb, T1, EB);

  dim3 gT(2, NGRAPH);
  k_bgemm128<<<gT, 256, 0, stream>>>(T1, T1, T2);
  k_bgemm128<<<gT, 256, 0, stream>>>(T2, T1, T3);
  k_bgemm128<<<gT, 256, 0, stream>>>(T3, T1, T4);

  k_rpe<<<(NGRAPH * NPG * NPG) / 256, 256, 0, stream>>>(
      T1, T2, T3, T4, rpe_W1, rpe_b1, rpe_W2, rpe_b2, EB, abias);

  const float scale = 1.0f / sqrtf((float)HD);

  for (int l = 0; l < NLAYER; ++l) {
    const float* Wq_l = Wq + (size_t)l * HID * HID;
    const float* Wk_l = Wk + (size_t)l * HID * HID;
    const float* Wv_l = Wv + (size_t)l * HID * HID;
    const float* Wo_l = Wo + (size_t)l * HID * HID;
    const float* W1_l = ffW1 + (size_t)l * HID * 2 * HID;
    const float* W2_l = ffW2 + (size_t)l * 2 * HID * HID;

    k_adddeg<<<NNODES, HID, 0, stream>>>(x, deg, degree_emb);

    int blk_qkv = (NNODES / 32) * (HID / 32) / 8;        // 64 blocks
    k_gemm<<<blk_qkv, 256, 0, stream>>>(x, Wq_l, bq + l * HID, q,  NNODES, HID, HID, 0);
    k_gemm<<<blk_qkv, 256, 0, stream>>>(x, Wk_l, bk + l * HID, kk, NNODES, HID, HID, 0);
    k_gemm<<<blk_qkv, 256, 0, stream>>>(x, Wv_l, bv + l * HID, vv, NNODES, HID, HID, 0);

    dim3 gS(2, NGRAPH * NH);
    k_scores<<<gS, 256, 0, stream>>>(q, kk, abias, sc, scale);
    k_softmax<<<(NGRAPH * NH * NPG) / 8, 256, 0, stream>>>(sc);
    dim3 gA(1, NGRAPH * NH);
    k_av<<<gA, 128, 0, stream>>>(sc, vv, att);

    k_gemm<<<blk_qkv, 256, 0, stream>>>(att, Wo_l, bo + l * HID, proj, NNODES, HID, HID, 0);
    k_ln<<<NNODES / 8, 256, 0, stream>>>(x, proj, ln1_s + l * HID, ln1_b + l * HID);

    int blk_ff1 = (NNODES / 32) * ((2 * HID) / 32) / 8;  // 128 blocks
    k_gemm<<<blk_ff1, 256, 0, stream>>>(x, W1_l, ffb1 + (size_t)l * 2 * HID, ffh,
                                        NNODES, HID, 2 * HID, 1);
    k_gemm<<<blk_qkv, 256, 0, stream>>>(ffh, W2_l, ffb2 + l * HID, proj,
                                        NNODES, 2 * HID, HID, 0);
    k_ln<<<NNODES / 8, 256, 0, stream>>>(x, proj, ln2_s + l * HID, ln2_b + l * HID);
  }

  k_pool<<<NGRAPH, HID, 0, stream>>>(x, (float*)d_out);
}